// GCNLayer_36524401885936
// MI455X (gfx1250) — compile-verified
//
#include <hip/hip_runtime.h>
#include <cmath>

#define N_NODES 100000
#define N_EDGES 800000
#define DIM_IN  128
#define DIM_OUT 256

typedef __attribute__((ext_vector_type(2))) float v2f;
typedef __attribute__((ext_vector_type(8))) float v8f;

// ---------------------------------------------------------------------------
// Kernel 1: zero the scatter accumulator m[N_NODES, DIM_IN] in workspace
// ---------------------------------------------------------------------------
__global__ void __launch_bounds__(256) zero_m(float4* __restrict__ m, int n4) {
    int i = blockIdx.x * blockDim.x + threadIdx.x;
    if (i < n4) m[i] = make_float4(0.f, 0.f, 0.f, 0.f);
}

// ---------------------------------------------------------------------------
// Kernel 2: m[dst] += x[src]   (wave32 per edge: 32 lanes x float4 = 128 f32)
// ---------------------------------------------------------------------------
__global__ void __launch_bounds__(256) scatter_add(
    const float* __restrict__ x,
    const int*   __restrict__ src,
    const int*   __restrict__ dst,
    float*       __restrict__ m)
{
    int edge = (blockIdx.x * blockDim.x + threadIdx.x) >> 5;   // wave id
    int lane = threadIdx.x & 31;
    if (edge >= N_EDGES) return;

    int s = src[edge];            // uniform per wave -> scalar load
    int d = dst[edge];

    const float4* xs = (const float4*)(x + (size_t)s * DIM_IN);
    float4 v = xs[lane];          // global_load_b128, 32 lanes cover 128 floats

    float* md = m + (size_t)d * DIM_IN + lane * 4;
    atomicAdd(md + 0, v.x);       // global_atomic_add_f32 (no return, agent)
    atomicAdd(md + 1, v.y);
    atomicAdd(md + 2, v.z);
    atomicAdd(md + 3, v.w);
}

// ---------------------------------------------------------------------------
// Kernel 3: h = gelu((m * deg_inv) @ W^T + b) via V_WMMA_F32_16X16X4_F32
//   block = 256 threads = 8 waves; block owns 32 rows x all 256 cols
//   wave w owns col-tiles [w*32, w*32+16) and [w*32+16, w*32+32), and both
//   16-row tiles -> each B fragment feeds 2 WMMAs (halves W load traffic)
// ---------------------------------------------------------------------------
#define A_STRIDE (DIM_IN + 4)   // 132 floats: A-frag reads hit distinct banks

__device__ __forceinline__ float gelu_exact(float v) {
    return 0.5f * v * (1.0f + erff(v * 0.70710678118654752440f));
}

__global__ void __launch_bounds__(256) gemm_bias_gelu(
    const float* __restrict__ m,
    const float* __restrict__ deg_inv,
    const float* __restrict__ W,      // [DIM_OUT, DIM_IN] row-major
    const float* __restrict__ bias,   // [DIM_OUT]
    float*       __restrict__ h)      // [N_NODES, DIM_OUT]
{
    __shared__ float a_tile[32 * A_STRIDE];

    const int tid  = threadIdx.x;
    const int row0 = blockIdx.x * 32;         // 3125 blocks, exact

    // ---- stage A tile (32 x 128), scaled by deg_inv, into LDS ----
    {
        int r = tid >> 3;                     // 0..31
        int c = (tid & 7) * 16;               // 0..112 step 16
        float scale = deg_inv[row0 + r];
        const float4* g = (const float4*)(m + (size_t)(row0 + r) * DIM_IN + c);
        float4* l = (float4*)(a_tile + r * A_STRIDE + c);
        #pragma unroll
        for (int i = 0; i < 4; ++i) {
            float4 v = g[i];
            v.x *= scale; v.y *= scale; v.z *= scale; v.w *= scale;
            l[i] = v;
        }
    }
    __syncthreads();

    const int lane  = tid & 31;
    const int wave  = tid >> 5;               // 0..7
    const int nloc  = lane & 15;              // N (and A-row M) position
    const int khalf = lane >> 4;              // selects K pair {0,1} vs {2,3}
    const int n0    = wave * 32;              // first column of this wave

    const float* arow0 = a_tile + nloc        * A_STRIDE;   // row tile 0
    const float* arow1 = a_tile + (16 + nloc) * A_STRIDE;   // row tile 1
    const float* w0    = W + (size_t)(n0 + nloc)      * DIM_IN;
    const float* w1    = W + (size_t)(n0 + 16 + nloc) * DIM_IN;

    v8f c00 = {};   // rows [0,16)  x cols [n0, n0+16)
    v8f c01 = {};   // rows [0,16)  x cols [n0+16, n0+32)
    v8f c10 = {};   // rows [16,32) x cols [n0, n0+16)
    v8f c11 = {};   // rows [16,32) x cols [n0+16, n0+32)

    #pragma unroll
    for (int kk = 0; kk < DIM_IN; kk += 4) {
        int k = kk + khalf * 2;
        v2f a0 = *(const v2f*)(arow0 + k);    // ds_load_b64 (pairs -> 2addr)
        v2f a1 = *(const v2f*)(arow1 + k);
        v2f b0 = *(const v2f*)(w0 + k);       // global_load_b64 (L2-hot W)
        v2f b1 = *(const v2f*)(w1 + k);
        c00 = __builtin_amdgcn_wmma_f32_16x16x4_f32(
                  false, a0, false, b0, (short)0, c00, false, false);
        c01 = __builtin_amdgcn_wmma_f32_16x16x4_f32(
                  false, a0, false, b1, (short)0, c01, false, false);
        c10 = __builtin_amdgcn_wmma_f32_16x16x4_f32(
                  false, a1, false, b0, (short)0, c10, false, false);
        c11 = __builtin_amdgcn_wmma_f32_16x16x4_f32(
                  false, a1, false, b1, (short)0, c11, false, false);
    }

    // ---- epilogue: bias + exact GELU + store ----
    // C/D layout: VGPR v -> row = khalf*8 + v, col = nloc
    float b0 = bias[n0 + nloc];
    float b1 = bias[n0 + 16 + nloc];
    size_t r0 = (size_t)row0 + khalf * 8;          // row tile 0 base
    size_t r1 = (size_t)row0 + 16 + khalf * 8;     // row tile 1 base

    #pragma unroll
    for (int v = 0; v < 8; ++v) {
        float* out0 = h + (r0 + v) * DIM_OUT;
        float* out1 = h + (r1 + v) * DIM_OUT;
        out0[n0 + nloc]      = gelu_exact(c00[v] + b0);
        out0[n0 + 16 + nloc] = gelu_exact(c01[v] + b1);
        out1[n0 + nloc]      = gelu_exact(c10[v] + b0);
        out1[n0 + 16 + nloc] = gelu_exact(c11[v] + b1);
    }
}

// ---------------------------------------------------------------------------
extern "C" void kernel_launch(void* const* d_in, const int* in_sizes, int n_in,
                              void* d_out, int out_size, void* d_ws, size_t ws_size,
                              hipStream_t stream) {
    (void)in_sizes; (void)n_in; (void)out_size; (void)ws_size;

    const float* x       = (const float*)d_in[0];
    const int*   edge    = (const int*)  d_in[1];   // [2, N_EDGES]
    const float* deg_inv = (const float*)d_in[2];
    const float* W       = (const float*)d_in[3];
    const float* bias    = (const float*)d_in[4];

    const int* src = edge;
    const int* dst = edge + N_EDGES;

    float* m = (float*)d_ws;                         // [N_NODES, DIM_IN]
    float* h = (float*)d_out;                        // [N_NODES, DIM_OUT]

    int n4 = N_NODES * DIM_IN / 4;
    zero_m<<<(n4 + 255) / 256, 256, 0, stream>>>((float4*)m, n4);

    int edge_blocks = (N_EDGES + 7) / 8;             // 8 waves per 256-thr block
    scatter_add<<<edge_blocks, 256, 0, stream>>>(x, src, dst, m);

    gemm_bias_gelu<<<N_NODES / 32, 256, 0, stream>>>(m, deg_inv, W, bias, h);
}